// GCN_40845138985205
// MI455X (gfx1250) — compile-verified
//
#include <hip/hip_runtime.h>
#include <math.h>

typedef float v2f __attribute__((ext_vector_type(2)));
typedef float v8f __attribute__((ext_vector_type(8)));

#define GCN_N 100000
#define GCN_E 1600000
#define GCN_INDIM 128
#define GCN_HID 64

// ---------------------------------------------------------------------------
// 1) per-edge degree accumulation (targets are 400KB -> L2 resident atomics)
// ---------------------------------------------------------------------------
__global__ __launch_bounds__(256) void gcn_deg_kernel(
    const int* __restrict__ src, const int* __restrict__ dst,
    float* __restrict__ deg_out, float* __restrict__ deg_in, int E) {
  int e = blockIdx.x * blockDim.x + threadIdx.x;
  if (e < E) {
    atomicAdd(&deg_out[src[e]], 1.0f);
    atomicAdd(&deg_in[dst[e]], 1.0f);
  }
}

// ---------------------------------------------------------------------------
// 2) norm = rsqrt(max(deg,1)) in place over both degree arrays at once
// ---------------------------------------------------------------------------
__global__ __launch_bounds__(256) void gcn_norm_kernel(float* __restrict__ d, int n) {
  int i = blockIdx.x * blockDim.x + threadIdx.x;
  if (i < n) d[i] = rsqrtf(fmaxf(d[i], 1.0f));
}

// ---------------------------------------------------------------------------
// 3) WMMA f32 GEMM:  X[N,64] = (A[N,K] * nsrc[N]) @ W[K,64]
//    One wave per 16-row tile.  V_WMMA_F32_16X16X4_F32 layouts (ISA 7.12.2):
//      A 16x4 : lanes0-15 row M=lane  v0=K0 v1=K1 ; lanes16-31 v0=K2 v1=K3
//      B 4x16 : lanes0-15 col N=lane  v0=K0 v1=K1 ; lanes16-31 v0=K2 v1=K3
//      C 16x16: vgpr i -> M = i + 8*(lane>=16), N = lane&15
// ---------------------------------------------------------------------------
__global__ __launch_bounds__(256) void gcn_gemm_wmma(
    const float* __restrict__ A, const float* __restrict__ W,
    const float* __restrict__ nsrc, float* __restrict__ X,
    int K, int nRows) {
  int wave = blockIdx.x * (blockDim.x >> 5) + (threadIdx.x >> 5);
  int lane = threadIdx.x & 31;
  int rowBlk = wave * 16;
  if (rowBlk >= nRows) return;

  int half = lane >> 4;      // 0 or 1
  int l    = lane & 15;
  int row  = rowBlk + l;
  float ns = nsrc[row];
  const float* arow = A + (long)row * K;

  v8f acc0 = {}, acc1 = {}, acc2 = {}, acc3 = {};

  for (int k0 = 0; k0 < K; k0 += 4) {
    int ka = k0 + half * 2;
    v2f a;
    a.x = arow[ka]     * ns;
    a.y = arow[ka + 1] * ns;
    const float* w0 = W + (long)ka * 64 + l;   // row ka
    const float* w1 = w0 + 64;                 // row ka+1
    v2f b;
    b.x = w0[0];  b.y = w1[0];
    acc0 = __builtin_amdgcn_wmma_f32_16x16x4_f32(false, a, false, b, (short)0, acc0, false, false);
    b.x = w0[16]; b.y = w1[16];
    acc1 = __builtin_amdgcn_wmma_f32_16x16x4_f32(false, a, false, b, (short)0, acc1, false, false);
    b.x = w0[32]; b.y = w1[32];
    acc2 = __builtin_amdgcn_wmma_f32_16x16x4_f32(false, a, false, b, (short)0, acc2, false, false);
    b.x = w0[48]; b.y = w1[48];
    acc3 = __builtin_amdgcn_wmma_f32_16x16x4_f32(false, a, false, b, (short)0, acc3, false, false);
  }

  float* xout = X + (long)rowBlk * 64;
#pragma unroll
  for (int i = 0; i < 8; ++i) {
    int r = i + half * 8;
    xout[(long)r * 64 +  0 + l] = acc0[i];
    xout[(long)r * 64 + 16 + l] = acc1[i];
    xout[(long)r * 64 + 32 + l] = acc2[i];
    xout[(long)r * 64 + 48 + l] = acc3[i];
  }
}

// ---------------------------------------------------------------------------
// 4) edge scatter-add: AGG[dst[e], :] += X[src[e], :]
//    16 lanes per edge, float4 per lane -> 256B coalesced gather per edge,
//    4x global_atomic_add_f32 scatter (agg buffer is L2 resident).
// ---------------------------------------------------------------------------
__global__ __launch_bounds__(256) void gcn_edge_agg(
    const int* __restrict__ src, const int* __restrict__ dst,
    const float* __restrict__ X, float* __restrict__ AGG, int E) {
  long tid = (long)blockIdx.x * blockDim.x + threadIdx.x;
  long e = tid >> 4;
  int  j = (int)(tid & 15) * 4;
  if (e < E) {
    int s = src[e];
    int d = dst[e];
    const float4 v = *(const float4*)(X + (long)s * 64 + j);
    float* p = AGG + (long)d * 64 + j;
    atomicAdd(p + 0, v.x);
    atomicAdd(p + 1, v.y);
    atomicAdd(p + 2, v.z);
    atomicAdd(p + 3, v.w);
  }
}

// ---------------------------------------------------------------------------
// 5) fused: x = elu(agg * ndst[node] + b[j])   (in place)
// ---------------------------------------------------------------------------
__global__ __launch_bounds__(256) void gcn_elu_kernel(
    float* __restrict__ AGG, const float* __restrict__ ndst,
    const float* __restrict__ b, int n) {
  int i = blockIdx.x * blockDim.x + threadIdx.x;
  if (i < n) {
    float v = AGG[i] * ndst[i >> 6] + b[i & 63];
    AGG[i] = v > 0.0f ? v : expm1f(v);
  }
}

// ---------------------------------------------------------------------------
// 6) final: out = out * ndst[node] + b2[j]   (in place on d_out)
// ---------------------------------------------------------------------------
__global__ __launch_bounds__(256) void gcn_final_kernel(
    float* __restrict__ OUT, const float* __restrict__ ndst,
    const float* __restrict__ b, int n) {
  int i = blockIdx.x * blockDim.x + threadIdx.x;
  if (i < n) OUT[i] = OUT[i] * ndst[i >> 6] + b[i & 63];
}

// ---------------------------------------------------------------------------
extern "C" void kernel_launch(void* const* d_in, const int* in_sizes, int n_in,
                              void* d_out, int out_size, void* d_ws, size_t ws_size,
                              hipStream_t stream) {
  const float* h   = (const float*)d_in[0];
  const int*   src = (const int*)  d_in[1];
  const int*   dst = (const int*)  d_in[2];
  const float* W1  = (const float*)d_in[3];
  const float* b1  = (const float*)d_in[4];
  const float* W2  = (const float*)d_in[5];
  const float* b2  = (const float*)d_in[6];
  float* out = (float*)d_out;

  const int N = in_sizes[0] / GCN_INDIM;   // 100000
  const int E = in_sizes[1];               // 1600000

  // workspace layout (floats): [norm_src N][norm_dst N][bufA N*64][bufB N*64]
  float* norm_src = (float*)d_ws;
  float* norm_dst = norm_src + N;
  float* bufA     = norm_dst + N;          // x1, later x2
  float* bufB     = bufA + (long)N * 64;   // agg1, then h2 (in place)

  const int feat = N * 64;

  // degrees -> norms
  hipMemsetAsync(norm_src, 0, (size_t)2 * N * sizeof(float), stream);
  gcn_deg_kernel<<<(E + 255) / 256, 256, 0, stream>>>(src, dst, norm_src, norm_dst, E);
  gcn_norm_kernel<<<(2 * N + 255) / 256, 256, 0, stream>>>(norm_src, 2 * N);

  // layer 1: project (WMMA), aggregate, fused norm+bias+ELU
  {
    int waves = N / 16;                       // N % 16 == 0
    int blocks = (waves + 7) / 8;             // 8 waves / 256-thread block
    gcn_gemm_wmma<<<blocks, 256, 0, stream>>>(h, W1, norm_src, bufA, GCN_INDIM, N);
  }
  hipMemsetAsync(bufB, 0, (size_t)feat * sizeof(float), stream);
  {
    long threads = (long)E * 16;
    gcn_edge_agg<<<(int)((threads + 255) / 256), 256, 0, stream>>>(src, dst, bufA, bufB, E);
  }
  gcn_elu_kernel<<<(feat + 255) / 256, 256, 0, stream>>>(bufB, norm_dst, b1, feat);

  // layer 2: project (WMMA), aggregate into d_out, fused norm+bias
  {
    int waves = N / 16;
    int blocks = (waves + 7) / 8;
    gcn_gemm_wmma<<<blocks, 256, 0, stream>>>(bufB, W2, norm_src, bufA, GCN_HID, N);
  }
  hipMemsetAsync(out, 0, (size_t)out_size * sizeof(float), stream);
  {
    long threads = (long)E * 16;
    gcn_edge_agg<<<(int)((threads + 255) / 256), 256, 0, stream>>>(src, dst, bufA, out, E);
  }
  gcn_final_kernel<<<(out_size + 255) / 256, 256, 0, stream>>>(out, norm_dst, b2, out_size);
}